// SimpleGCN_39702677684790
// MI455X (gfx1250) — compile-verified
//
#include <hip/hip_runtime.h>
#include <hip/hip_bf16.h>
#include <math.h>

#define HID   64
#define INDIM 128
#define EMB   32
#define EPSV  1e-5f

typedef __attribute__((ext_vector_type(2))) float v2f;
typedef __attribute__((ext_vector_type(8))) float v8f;

// ---------------------------------------------------------------------------
// C[rows x 64] = A[rows x K] * B[K x 64], fp32 WMMA 16x16x4.
// One wave per 16x16 output tile. rows must be multiple of 16 (100000 is).
// A frag (16x4 f32): lane = m + 16*kh ; v0 = A[m][2kh], v1 = A[m][2kh+1]
// B frag (4x16 f32): lane = n + 16*kh ; v0 = B[2kh][n], v1 = B[2kh+1][n]
// C/D (16x16 f32):  vgpr r, lane l -> row = r + 8*(l>>4), col = l&15
// ---------------------------------------------------------------------------
__global__ __launch_bounds__(256) void gemm_f32_wmma(
    const float* __restrict__ A, const float* __restrict__ B,
    float* __restrict__ C, int rows, int K) {
  int wave = (int)((blockIdx.x * blockDim.x + threadIdx.x) >> 5);
  int lane = threadIdx.x & 31;
  int rt = wave >> 2;   // row tile
  int ct = wave & 3;    // col tile (64 / 16)
  if (rt * 16 >= rows) return;
  int m  = lane & 15;   // row (A) / col (B)
  int kh = lane >> 4;   // k-half select
  const float* arow = A + (size_t)(rt * 16 + m) * K;
  const float* bcol = B + (ct * 16 + m);
  v8f acc = {};
  for (int k0 = 0; k0 < K; k0 += 4) {
    v2f a, b;
    a.x = arow[k0 + 2 * kh + 0];
    a.y = arow[k0 + 2 * kh + 1];
    b.x = bcol[(size_t)(k0 + 2 * kh + 0) * HID];
    b.y = bcol[(size_t)(k0 + 2 * kh + 1) * HID];
    acc = __builtin_amdgcn_wmma_f32_16x16x4_f32(
        false, a, false, b, (short)0, acc, false, false);
  }
#pragma unroll
  for (int r = 0; r < 8; ++r) {
    int row = rt * 16 + r + 8 * kh;
    C[(size_t)row * HID + ct * 16 + m] = acc[r];
  }
}

// ---------------------------------------------------------------------------
__global__ void deg_accum(const int* __restrict__ dst, float* __restrict__ deg,
                          int E) {
  int e = blockIdx.x * blockDim.x + threadIdx.x;
  if (e < E) atomicAdd(&deg[dst[e]], 1.0f);
}

__global__ void deg_finalize(float* __restrict__ deg, float* __restrict__ dinv,
                             int N) {
  int i = blockIdx.x * blockDim.x + threadIdx.x;
  if (i < N) {
    float d = deg[i] + 1.0f;   // self loop
    deg[i]  = d;
    dinv[i] = rsqrtf(d);
  }
}

// One wave per edge; each lane handles 2 features. Atomics land in L2.
__global__ __launch_bounds__(256) void scatter_edges(
    const int* __restrict__ src, const int* __restrict__ dst,
    const float* __restrict__ dinv, const float* __restrict__ h,
    float* __restrict__ agg, int E) {
  int wave = (int)((blockIdx.x * blockDim.x + threadIdx.x) >> 5);
  int lane = threadIdx.x & 31;
  if (wave >= E) return;
  int s = src[wave], d = dst[wave];
  float nrm = dinv[s] * dinv[d];
  const float2* hp = (const float2*)(h + (size_t)s * HID);
  float2 v = hp[lane];
  float* out = agg + (size_t)d * HID + lane * 2;
  atomicAdd(out + 0, v.x * nrm);
  atomicAdd(out + 1, v.y * nrm);
}

// y = relu(agg + h/deg + bias); in-place over agg; per-feature sums reduced
// in LDS per 64-node block, one global atomic per feature per block.
__global__ __launch_bounds__(256) void selfloop_relu_fsum(
    const float* __restrict__ h, const float* __restrict__ deg,
    const float* __restrict__ bias, float* __restrict__ y,
    float* __restrict__ fsum, int N) {
  __shared__ float sh[HID];
  int t = threadIdx.x;
  if (t < HID) sh[t] = 0.f;
  __syncthreads();
  int fp = t & 31;   // feature pair
  int r  = t >> 5;   // 0..7
  int base = blockIdx.x * 64;
  float2 acc = {0.f, 0.f};
  float b0 = bias[2 * fp], b1 = bias[2 * fp + 1];
  for (int j = 0; j < 8; ++j) {
    int i = base + r + j * 8;
    if (i < N) {
      float inv = 1.0f / deg[i];
      const float2* hp = (const float2*)(h + (size_t)i * HID);
      float2* yp = (float2*)(y + (size_t)i * HID);
      float2 a  = yp[fp];
      float2 hv = hp[fp];
      float y0 = fmaxf(a.x + hv.x * inv + b0, 0.f);
      float y1 = fmaxf(a.y + hv.y * inv + b1, 0.f);
      yp[fp] = make_float2(y0, y1);
      acc.x += y0;
      acc.y += y1;
    }
  }
  atomicAdd(&sh[2 * fp + 0], acc.x);
  atomicAdd(&sh[2 * fp + 1], acc.y);
  __syncthreads();
  if (t < HID) atomicAdd(&fsum[t], sh[t]);
}

__global__ void compute_mean(const float* __restrict__ fsum,
                             float* __restrict__ mean, float invN) {
  int f = threadIdx.x;
  mean[f] = fsum[f] * invN;
}

// xc = y - ms*mean (in place); accumulate per-feature sum(xc^2)
__global__ __launch_bounds__(256) void center_sumsq(
    float* __restrict__ y, const float* __restrict__ mean,
    const float* __restrict__ ms, float* __restrict__ fsq, int N) {
  __shared__ float sh[HID];
  int t = threadIdx.x;
  if (t < HID) sh[t] = 0.f;
  __syncthreads();
  int fp = t & 31;
  int r  = t >> 5;
  int base = blockIdx.x * 64;
  float2 acc = {0.f, 0.f};
  float m0 = ms[2 * fp] * mean[2 * fp];
  float m1 = ms[2 * fp + 1] * mean[2 * fp + 1];
  for (int j = 0; j < 8; ++j) {
    int i = base + r + j * 8;
    if (i < N) {
      float2* yp = (float2*)(y + (size_t)i * HID);
      float2 v = yp[fp];
      v.x -= m0;
      v.y -= m1;
      yp[fp] = v;
      acc.x += v.x * v.x;
      acc.y += v.y * v.y;
    }
  }
  atomicAdd(&sh[2 * fp + 0], acc.x);
  atomicAdd(&sh[2 * fp + 1], acc.y);
  __syncthreads();
  if (t < HID) atomicAdd(&fsq[t], sh[t]);
}

__global__ void compute_gscale(const float* __restrict__ fsq,
                               const float* __restrict__ gw,
                               float* __restrict__ gscale, float invN) {
  int f = threadIdx.x;
  gscale[f] = gw[f] * rsqrtf(fsq[f] * invN + EPSV);
}

// t = gscale*xc + gb, then LayerNorm over the 64 features of each node.
// One wave per node (wave32: 2 features/lane, shfl_xor reductions).
__global__ __launch_bounds__(256) void gnorm_lnorm(
    const float* __restrict__ xc, const float* __restrict__ gscale,
    const float* __restrict__ gb, const float* __restrict__ lw,
    const float* __restrict__ lb, float* __restrict__ out, int N) {
  int wave = (int)((blockIdx.x * blockDim.x + threadIdx.x) >> 5);
  int lane = threadIdx.x & 31;
  if (wave >= N) return;
  const float2* xp = (const float2*)(xc + (size_t)wave * HID);
  float2 x = xp[lane];
  float t0 = gscale[2 * lane] * x.x + gb[2 * lane];
  float t1 = gscale[2 * lane + 1] * x.y + gb[2 * lane + 1];
  float s = t0 + t1;
  for (int off = 16; off; off >>= 1) s += __shfl_xor(s, off);
  float mean = s * (1.0f / HID);
  float d0 = t0 - mean, d1 = t1 - mean;
  float q = d0 * d0 + d1 * d1;
  for (int off = 16; off; off >>= 1) q += __shfl_xor(q, off);
  float rstd = rsqrtf(q * (1.0f / HID) + EPSV);
  float2 o;
  o.x = lw[2 * lane] * d0 * rstd + lb[2 * lane];
  o.y = lw[2 * lane + 1] * d1 * rstd + lb[2 * lane + 1];
  ((float2*)(out + (size_t)wave * HID))[lane] = o;
}

// ---------------------------------------------------------------------------
__global__ void init_neg_inf(float* __restrict__ p, int n) {
  int i = blockIdx.x * blockDim.x + threadIdx.x;
  if (i < n) p[i] = -__builtin_inff();
}

__device__ inline void atomicMaxFloat(float* addr, float val) {
  int* ia = (int*)addr;
  int old = __float_as_int(*addr);
  while (val > __int_as_float(old)) {
    int prev = atomicCAS(ia, old, __float_as_int(val));
    if (prev == old) break;
    old = prev;
  }
}

__global__ __launch_bounds__(256) void maxpool(const float* __restrict__ h,
                                               float* __restrict__ pooled,
                                               int N) {
  __shared__ float sd[256];
  int t = threadIdx.x;
  int f = t & 63, r = t >> 6;  // r in 0..3
  float m = -__builtin_inff();
  const int CH = 512;
  int base = blockIdx.x * CH;
  for (int j = r; j < CH; j += 4) {
    int i = base + j;
    if (i < N) m = fmaxf(m, h[(size_t)i * HID + f]);
  }
  sd[t] = m;
  __syncthreads();
  if (r == 0) {
    m = fmaxf(fmaxf(sd[f], sd[f + 64]), fmaxf(sd[f + 128], sd[f + 192]));
    atomicMaxFloat(&pooled[f], m);
  }
}

__global__ void fc_kernel(const float* __restrict__ pooled,
                          const float* __restrict__ W,
                          const float* __restrict__ b,
                          float* __restrict__ out) {
  int j = threadIdx.x;  // 0..31
  float s = b[j];
  for (int k = 0; k < HID; ++k) s += pooled[k] * W[k * EMB + j];
  out[j] = s;
}

// ---------------------------------------------------------------------------
extern "C" void kernel_launch(void* const* d_in, const int* in_sizes, int n_in,
                              void* d_out, int out_size, void* d_ws,
                              size_t ws_size, hipStream_t stream) {
  const float* x    = (const float*)d_in[0];
  const int*   ei   = (const int*)d_in[1];
  const float* W1   = (const float*)d_in[2];
  const float* b1   = (const float*)d_in[3];
  const float* W2   = (const float*)d_in[4];
  const float* b2   = (const float*)d_in[5];
  const float* gnw  = (const float*)d_in[6];
  const float* gnb  = (const float*)d_in[7];
  const float* gnms = (const float*)d_in[8];
  const float* lnw  = (const float*)d_in[9];
  const float* lnb  = (const float*)d_in[10];
  const float* fcW  = (const float*)d_in[11];
  const float* fcb  = (const float*)d_in[12];

  int N = in_sizes[0] / INDIM;   // 100000
  int E = in_sizes[1] / 2;       // 1600000
  const int* src = ei;
  const int* dst = ei + E;

  float* buf0   = (float*)d_ws;                  // N*64
  float* buf1   = buf0 + (size_t)N * HID;        // N*64
  float* deg    = buf1 + (size_t)N * HID;        // N
  float* dinv   = deg + N;                       // N
  float* fsum   = dinv + N;                      // 64
  float* mean   = fsum + HID;                    // 64
  float* fsq    = mean + HID;                    // 64
  float* gscale = fsq + HID;                     // 64
  float* pooled = gscale + HID;                  // 64

  float invN = 1.0f / (float)N;
  int rowTiles = (N + 15) / 16;
  int gemmBlocks = (rowTiles * 4 + 7) / 8;       // 8 waves / block

  // degrees (shared by both layers)
  hipMemsetAsync(deg, 0, (size_t)N * sizeof(float), stream);
  deg_accum<<<(E + 255) / 256, 256, 0, stream>>>(dst, deg, E);
  deg_finalize<<<(N + 255) / 256, 256, 0, stream>>>(deg, dinv, N);

  // ---- layer 1 ----
  gemm_f32_wmma<<<gemmBlocks, 256, 0, stream>>>(x, W1, buf0, N, INDIM);
  hipMemsetAsync(buf1, 0, (size_t)N * HID * sizeof(float), stream);
  scatter_edges<<<(E + 7) / 8, 256, 0, stream>>>(src, dst, dinv, buf0, buf1, E);
  hipMemsetAsync(fsum, 0, HID * sizeof(float), stream);
  selfloop_relu_fsum<<<(N + 63) / 64, 256, 0, stream>>>(buf0, deg, b1, buf1,
                                                        fsum, N);
  compute_mean<<<1, HID, 0, stream>>>(fsum, mean, invN);
  hipMemsetAsync(fsq, 0, HID * sizeof(float), stream);
  center_sumsq<<<(N + 63) / 64, 256, 0, stream>>>(buf1, mean, gnms, fsq, N);
  compute_gscale<<<1, HID, 0, stream>>>(fsq, gnw, gscale, invN);
  gnorm_lnorm<<<(N + 7) / 8, 256, 0, stream>>>(buf1, gscale, gnb, lnw, lnb,
                                               buf0, N);

  // ---- layer 2 ----
  gemm_f32_wmma<<<gemmBlocks, 256, 0, stream>>>(buf0, W2, buf1, N, HID);
  hipMemsetAsync(buf0, 0, (size_t)N * HID * sizeof(float), stream);
  scatter_edges<<<(E + 7) / 8, 256, 0, stream>>>(src, dst, dinv, buf1, buf0, E);
  hipMemsetAsync(fsum, 0, HID * sizeof(float), stream);
  selfloop_relu_fsum<<<(N + 63) / 64, 256, 0, stream>>>(buf1, deg, b2, buf0,
                                                        fsum, N);
  compute_mean<<<1, HID, 0, stream>>>(fsum, mean, invN);
  hipMemsetAsync(fsq, 0, HID * sizeof(float), stream);
  center_sumsq<<<(N + 63) / 64, 256, 0, stream>>>(buf0, mean, gnms, fsq, N);
  compute_gscale<<<1, HID, 0, stream>>>(fsq, gnw, gscale, invN);
  gnorm_lnorm<<<(N + 7) / 8, 256, 0, stream>>>(buf0, gscale, gnb, lnw, lnb,
                                               buf1, N);

  // ---- pool + fc ----
  init_neg_inf<<<1, HID, 0, stream>>>(pooled, HID);
  maxpool<<<(N + 511) / 512, 256, 0, stream>>>(buf1, pooled, N);
  fc_kernel<<<1, EMB, 0, stream>>>(pooled, fcW, fcb, (float*)d_out);
}